// BahdanauAttention_softmax_30064771072921
// MI455X (gfx1250) — compile-verified
//
#include <hip/hip_runtime.h>
#include <hip/hip_bf16.h>

typedef float v2f __attribute__((ext_vector_type(2)));
typedef float v8f __attribute__((ext_vector_type(8)));

#define BATCH 256
#define NQ    16
#define KK    200
#define HID   128
#define KSZ   64
#define NEG_INF_SCALE (-1000000000.0f)

// CDNA5 hardware tanh (TRANS pipe). Falls back to inline asm if the builtin
// name differs on this toolchain.
__device__ __forceinline__ float fast_tanh(float x) {
#if __has_builtin(__builtin_amdgcn_tanhf)
    return __builtin_amdgcn_tanhf(x);
#else
    float y;
    asm volatile("v_tanh_f32 %0, %1\n\ts_delay_alu instid0(TRANS32_DEP_1)"
                 : "=v"(y) : "v"(x));
    return y;
#endif
}

// -----------------------------------------------------------------------------
// Projection GEMM: Out(M,128) = A(M,64) @ W(64,128) + bias(128)
// One wave32 per 16x16 output tile, V_WMMA_F32_16X16X4_F32 over K=64 (16 steps).
// Exact f32 — same numerics class as the reference.
// -----------------------------------------------------------------------------
__global__ __launch_bounds__(256) void proj_gemm_wmma(
    const float* __restrict__ A, const float* __restrict__ W,
    const float* __restrict__ bias, float* __restrict__ Out, int M) {
    const int waveId = (blockIdx.x * blockDim.x + threadIdx.x) >> 5;
    const int lane   = threadIdx.x & 31;
    const int mt = waveId >> 3;          // 8 N-tiles (128/16)
    const int nt = waveId & 7;
    if (mt * 16 >= M) return;
    const int m0 = mt * 16, n0 = nt * 16;
    const int half = lane >> 4;          // 0: lanes 0-15, 1: lanes 16-31
    const int lm   = lane & 15;

    v8f acc;
    const float bn = bias[n0 + lm];      // bias broadcast along rows
#pragma unroll
    for (int r = 0; r < 8; ++r) acc[r] = bn;

    const float* arow = A + (size_t)(m0 + lm) * KSZ;
#pragma unroll
    for (int k0 = 0; k0 < KSZ; k0 += 4) {
        v2f a, b;
        // A fragment 16x4: lanes 0-15 -> K = k0,k0+1 ; lanes 16-31 -> K = k0+2,k0+3
        a.x = arow[k0 + 2 * half];
        a.y = arow[k0 + 2 * half + 1];
        // B fragment 4x16: row-striped across lanes within a VGPR
        b.x = W[(size_t)(k0 + 2 * half) * HID + n0 + lm];
        b.y = W[(size_t)(k0 + 2 * half + 1) * HID + n0 + lm];
        acc = __builtin_amdgcn_wmma_f32_16x16x4_f32(false, a, false, b,
                                                    (short)0, acc, false, false);
    }
#pragma unroll
    for (int r = 0; r < 8; ++r)
        Out[(size_t)(m0 + r + 8 * half) * HID + n0 + lm] = acc[r];
}

// -----------------------------------------------------------------------------
// Fused scores -> softmax -> attention-output, one workgroup (8 waves) per batch.
// Avoids the 419 MB (B,NQ,K,H) intermediate entirely.
// -----------------------------------------------------------------------------
__global__ __launch_bounds__(256) void attn_fused_kernel(
    const float* __restrict__ qh,    const float* __restrict__ kh,
    const float* __restrict__ keys,  const float* __restrict__ masks,
    const float* __restrict__ Wv,    const float* __restrict__ bv,
    float* __restrict__ out_attn,    float* __restrict__ out_w) {
    const int b   = blockIdx.x;
    const int tid = threadIdx.x;

    __shared__ float qh_s[NQ * HID];     // 8 KB
    __shared__ float wv_s[HID];          // 0.5 KB
    __shared__ float sc_s[NQ * KK];      // 12.5 KB: scores, then weights

    const float* qhb = qh + (size_t)b * NQ * HID;
    for (int i = tid; i < NQ * HID; i += 256) qh_s[i] = qhb[i];
    for (int i = tid; i < HID; i += 256)      wv_s[i] = Wv[i];
    __syncthreads();

    // ---- Phase 1: scores[q][k] = bv + sum_h Wv[h]*tanh(qh[q][h]+kh[k][h]) + mask
    const float  bvv   = bv[0];
    const float* khb   = kh    + (size_t)b * KK * HID;   // hot in 192MB L2
    const float* maskb = masks + (size_t)b * NQ * KK;
    for (int p = tid; p < NQ * KK; p += 256) {
        const int q = p / KK;
        const int k = p - q * KK;
        const float* qrow = qh_s + q * HID;
        const float* krow = khb + (size_t)k * HID;
        float acc = 0.f;
#pragma unroll 4
        for (int h = 0; h < HID; h += 4) {
            float4 qv = *(const float4*)(qrow + h);
            float4 kv = *(const float4*)(krow + h);
            float4 wv = *(const float4*)(wv_s + h);
            acc += wv.x * fast_tanh(qv.x + kv.x);
            acc += wv.y * fast_tanh(qv.y + kv.y);
            acc += wv.z * fast_tanh(qv.z + kv.z);
            acc += wv.w * fast_tanh(qv.w + kv.w);
        }
        sc_s[p] = acc + bvv + maskb[p] * NEG_INF_SCALE;
    }
    __syncthreads();

    // ---- Phase 2: softmax over K=200 per row; 8 waves x 2 rows, wave32 shuffles
    const int wave = tid >> 5;
    const int lane = tid & 31;
    for (int q = wave; q < NQ; q += 8) {
        float* row = sc_s + q * KK;
        float m = -3.402823466e38f;
        for (int k = lane; k < KK; k += 32) m = fmaxf(m, row[k]);
#pragma unroll
        for (int off = 16; off > 0; off >>= 1) m = fmaxf(m, __shfl_xor(m, off, 32));
        float s = 0.f;
        for (int k = lane; k < KK; k += 32) {
            float e = __expf(row[k] - m);
            row[k] = e;
            s += e;
        }
#pragma unroll
        for (int off = 16; off > 0; off >>= 1) s += __shfl_xor(s, off, 32);
        const float inv = 1.0f / s;
        float* wout = out_w + ((size_t)b * NQ + q) * KK;
        for (int k = lane; k < KK; k += 32) {
            float wv = row[k] * inv;
            row[k]  = wv;        // keep normalized weights in LDS for phase 3
            wout[k] = wv;        // emit attn_weight (B,NQ,K,1)
        }
    }
    __syncthreads();

    // ---- Phase 3: attn_out(16x64) = weights(16x200) @ keys(200x64) via WMMA f32.
    // Waves 0..3 each own one 16-wide N-tile; 50 WMMA steps over K=200.
    if (wave < 4) {
        const int n0   = wave * 16;
        const int half = lane >> 4;
        const int lm   = lane & 15;
        v8f acc = {};
        const float* keyb = keys + (size_t)b * KK * KSZ;
        for (int k0 = 0; k0 < KK; k0 += 4) {
            v2f a, bf;
            a.x  = sc_s[lm * KK + k0 + 2 * half];
            a.y  = sc_s[lm * KK + k0 + 2 * half + 1];
            bf.x = keyb[(size_t)(k0 + 2 * half) * KSZ + n0 + lm];
            bf.y = keyb[(size_t)(k0 + 2 * half + 1) * KSZ + n0 + lm];
            acc = __builtin_amdgcn_wmma_f32_16x16x4_f32(false, a, false, bf,
                                                        (short)0, acc, false, false);
        }
        float* outb = out_attn + (size_t)b * NQ * KSZ;
#pragma unroll
        for (int r = 0; r < 8; ++r)
            outb[(size_t)(r + 8 * half) * KSZ + n0 + lm] = acc[r];
    }
}

extern "C" void kernel_launch(void* const* d_in, const int* in_sizes, int n_in,
                              void* d_out, int out_size, void* d_ws, size_t ws_size,
                              hipStream_t stream) {
    const float* queries = (const float*)d_in[0];  // (256,16,64)
    const float* keys    = (const float*)d_in[1];  // (256,200,64)
    const float* masks   = (const float*)d_in[2];  // (256,16,200,1)
    // d_in[3] = num_neg (unused)
    const float* Wq = (const float*)d_in[4];       // (64,128)
    const float* bq = (const float*)d_in[5];       // (128)
    const float* Wk = (const float*)d_in[6];       // (64,128)
    const float* bk = (const float*)d_in[7];       // (128)
    const float* Wv = (const float*)d_in[8];       // (128,1)
    const float* bv = (const float*)d_in[9];       // (1)

    float* ws = (float*)d_ws;
    float* qh = ws;                                  // 4096  x 128 (2 MB)
    float* kh = ws + (size_t)BATCH * NQ * HID;       // 51200 x 128 (26 MB, L2-resident)

    float* out_attn = (float*)d_out;                              // 256*16*64
    float* out_w    = (float*)d_out + (size_t)BATCH * NQ * KSZ;   // 256*16*200

    // qh: M=4096 -> 256 m-tiles * 8 n-tiles = 2048 waves = 256 blocks of 8 waves
    proj_gemm_wmma<<<256, 256, 0, stream>>>(queries, Wq, bq, qh, BATCH * NQ);
    // kh: M=51200 -> 3200 m-tiles * 8 n-tiles = 25600 waves = 3200 blocks
    proj_gemm_wmma<<<3200, 256, 0, stream>>>(keys, Wk, bk, kh, BATCH * KK);
    // fused scores/softmax/output: one workgroup per batch
    attn_fused_kernel<<<BATCH, 256, 0, stream>>>(qh, kh, keys, masks, Wv, bv,
                                                 out_attn, out_w);
}